// ClimbGenerator_60808146977437
// MI455X (gfx1250) — compile-verified
//
#include <hip/hip_runtime.h>
#include <hip/hip_bf16.h>

// ---------------------------------------------------------------------------
// MI455X / gfx1250: GraphConv-GNN (10 layers) + 396x396 MLP (3 layers) + softmax.
//
// Roofline: scatter/segment-sum dominates (~15-20 GB total traffic; the
// 162 MB feature buffer fits the 192 MB L2 so gathers mostly hit L2).
// Dense transforms (~44 GFLOP total) use V_WMMA_F32_16X16X4_F32 (full fp32).
//
// Key codegen fix vs previous round: the WMMA K-loop is guard-free.
//  * All intermediate feature buffers use K-strides padded to a multiple of 4
//    with *zero* padding, so A-fragments are unguarded global_load_b64.
//  * Each block stages its 16-column weight slice (zero-padded) + bias into
//    LDS, so B-fragments are unguarded ds_load_b64 reused by all 8 waves.
//  * N row-tiles are exact (811008 and 4096 are multiples of 16).
//  * Zero-padded B columns make padded output columns exactly 0, which feeds
//    the next layer's padded-K loads with correct zeros for free.
// ---------------------------------------------------------------------------

typedef float v2f __attribute__((ext_vector_type(2)));
typedef float v8f __attribute__((ext_vector_type(8)));

// --------------------------- pad x (N x 5 -> N x 8, zero-filled) -----------
__global__ __launch_bounds__(256) void pad_x(
    const float* __restrict__ x, float* __restrict__ xp, long long N)
{
    long long t = (long long)blockIdx.x * 256 + threadIdx.x;   // over N*8
    if (t >= N * 8) return;
    long long n = t >> 3;
    int k = (int)(t & 7);
    xp[t] = (k < 5) ? x[n * 5 + k] : 0.0f;
}

// --------------------------- scatter / segment-sum -------------------------
// msg = h[src] * w ; agg[dst] += msg   (thread per (edge, channel), stride s)
__global__ __launch_bounds__(256) void scatter_edges(
    const float* __restrict__ h,
    const long long* __restrict__ src,
    const long long* __restrict__ dst,
    const float* __restrict__ w,
    float* __restrict__ agg,
    long long E, int fin, int s)
{
    long long t = (long long)blockIdx.x * 256 + threadIdx.x;
    long long total = E * (long long)fin;
    if (t >= total) return;
    long long e = t / fin;
    int c = (int)(t - e * (long long)fin);
    long long si = src[e];
    long long di = dst[e];
    float m = h[si * (long long)s + c] * w[e];
    atomicAdd(&agg[di * (long long)s + c], m);
}

// ------------------------------ WMMA GEMM ----------------------------------
// Out[n,m] = act( A1[n,:]·W1[m,:] (+ A2[n,:]·W2[m,:]) + bias[m] )
// A1/A2: N x K, row stride lda (lda = pad4(K), padding zero-filled).
// W1/W2: M x K row-major (global); staged per-block into LDS, zero-padded to
//        16 x Kp so the inner loop needs no guards.
// Out:   N x *, row stride ldo; columns [M, min(16-tile, ldo)) receive exact
//        zeros (B padding ⇒ zero accumulator ⇒ zero store).
//
// Grid: blockIdx.x = rowGroup * colTiles + colTile; 8 waves per block, each
// wave computes one 16x16 tile: (rowGroup*8 + waveId, colTile).
//
// Fragment layouts per CDNA5 ISA (CDNA5 ISA, 16x16x4 f32):
//   A: lanes 0-15 row=lane  {v0:K+0, v1:K+1}; lanes 16-31 row=lane-16 {K+2,K+3}
//   B: lanes 0-15 col=lane  {v0:K+0, v1:K+1}; lanes 16-31 col=lane-16 {K+2,K+3}
//   C: VGPR r -> row (r + 8*half), col = lane&15
// EXEC all-ones around WMMA: only wave-uniform exits, no per-lane branches.
__global__ __launch_bounds__(256) void gemm_dual_wmma(
    const float* __restrict__ A1, const float* __restrict__ W1g,
    const float* __restrict__ A2, const float* __restrict__ W2g,
    const float* __restrict__ biasg,
    float* __restrict__ Out,
    int N, int M, int K, int lda, int ldo, int colTiles, int relu)
{
    extern __shared__ float smem[];
    const int Kp = (K + 3) & ~3;

    const int colTile  = blockIdx.x % colTiles;
    const int rowGroup = blockIdx.x / colTiles;
    const int colBase  = colTile << 4;

    float* W1s = smem;                       // 16 * Kp
    float* W2s = smem + 16 * Kp;             // 16 * Kp (only if A2)
    float* bs  = smem + 16 * Kp * (A2 ? 2 : 1);  // 16

    // ---- cooperative zero-padded staging of this block's weight slice ----
    for (int i = threadIdx.x; i < 16 * Kp; i += 256) {
        int mi = i / Kp;
        int k  = i - mi * Kp;
        int col = colBase + mi;
        bool ok = (col < M) && (k < K);
        W1s[i] = ok ? W1g[(long long)col * K + k] : 0.0f;
        if (A2) W2s[i] = ok ? W2g[(long long)col * K + k] : 0.0f;
    }
    if (threadIdx.x < 16) {
        int col = colBase + threadIdx.x;
        bs[threadIdx.x] = (col < M) ? biasg[col] : 0.0f;
    }
    __syncthreads();

    const int waveId  = threadIdx.x >> 5;
    const int lane    = threadIdx.x & 31;
    const int rowTile = rowGroup * 8 + waveId;
    const int rowBase = rowTile << 4;
    if (rowBase >= N) return;                // wave-uniform exit

    const int half = lane >> 4;              // 0: lanes 0-15, 1: lanes 16-31
    const int lr   = lane & 15;

    const float* a1p = A1 + (long long)(rowBase + lr) * lda + half * 2;
    const float* b1p = W1s + lr * Kp + half * 2;

    v8f c = {};

    #pragma unroll 2
    for (int k = 0; k < Kp; k += 4) {
        v2f a = *(const v2f*)(a1p + k);      // global_load_b64, unguarded
        v2f b = *(const v2f*)(b1p + k);      // ds_load_b64, unguarded
        c = __builtin_amdgcn_wmma_f32_16x16x4_f32(
                false, a, false, b, (short)0, c, false, false);
    }

    if (A2 != nullptr) {
        const float* a2p = A2 + (long long)(rowBase + lr) * lda + half * 2;
        const float* b2p = W2s + lr * Kp + half * 2;
        #pragma unroll 2
        for (int k = 0; k < Kp; k += 4) {
            v2f a = *(const v2f*)(a2p + k);
            v2f b = *(const v2f*)(b2p + k);
            c = __builtin_amdgcn_wmma_f32_16x16x4_f32(
                    false, a, false, b, (short)0, c, false, false);
        }
    }

    // ---- epilogue: bias + optional ReLU; padded cols are exact zeros ----
    const int col = colBase + lr;
    const float bv = bs[lr];
    #pragma unroll
    for (int r = 0; r < 8; ++r) {
        int row = rowBase + r + half * 8;
        float v = c[r] + bv;
        if (relu) v = fmaxf(v, 0.0f);
        if (col < ldo)
            Out[(long long)row * ldo + col] = v;
    }
}

// ------------------------------- softmax(2) --------------------------------
__global__ __launch_bounds__(256) void softmax2(
    const float* __restrict__ z, float* __restrict__ out, long long P)
{
    long long i = (long long)blockIdx.x * 256 + threadIdx.x;
    if (i >= P) return;
    float a = z[2 * i];
    float b = z[2 * i + 1];
    float m  = fmaxf(a, b);
    float ea = expf(a - m);
    float eb = expf(b - m);
    float inv = 1.0f / (ea + eb);
    out[2 * i]     = ea * inv;
    out[2 * i + 1] = eb * inv;
}

// ------------------------------- host driver -------------------------------
extern "C" void kernel_launch(void* const* d_in, const int* in_sizes, int n_in,
                              void* d_out, int out_size, void* d_ws, size_t ws_size,
                              hipStream_t stream)
{
    static const int FIN [10] = {5, 50, 50, 50, 50, 50, 20, 15, 10,  5};
    static const int FOUT[10] = {50, 50, 50, 50, 50, 20, 15, 10,  5,  2};
    const int MLP_DIM = 396;
    auto pad4 = [](int k) { return (k + 3) & ~3; };

    const float*     x  = (const float*)d_in[0];
    const long long* ei = (const long long*)d_in[1]; // (2, E) int64
    const float*     ew = (const float*)d_in[2];

    long long N = (long long)in_sizes[0] / 5;   // 811008 (multiple of 16)
    long long E = (long long)in_sizes[2];       // 6488064
    const long long* src = ei;
    const long long* dst = ei + E;

    // workspace layout (floats): [agg: N*52][hA: N*52][hB: N*52]
    float* ws  = (float*)d_ws;
    float* agg = ws;
    float* hA  = ws + N * 52;
    float* hB  = ws + 2 * N * 52;

    // Pad x (N x 5, stride 5) -> xpad (N x 8, zero-padded). xpad lives in hB,
    // which is only written again at layer 1 (after xpad is dead).
    float* xpad = hB;
    {
        long long total = N * 8;
        unsigned blocks = (unsigned)((total + 255) / 256);
        pad_x<<<blocks, 256, 0, stream>>>(x, xpad, N);
    }

    const float* h = xpad;

    // ---- 10 GraphConv layers ----
    for (int l = 0; l < 10; ++l) {
        int fin  = FIN[l], fout = FOUT[l];
        int sIn  = pad4(fin);                       // stride of h and agg
        int sOut = (l == 9) ? 2 : pad4(fout);       // last layer compact for reshape

        hipMemsetAsync(agg, 0, (size_t)N * sIn * sizeof(float), stream);

        long long total = E * (long long)fin;
        unsigned sblocks = (unsigned)((total + 255) / 256);
        scatter_edges<<<sblocks, 256, 0, stream>>>(h, src, dst, ew, agg, E, fin, sIn);

        const float* Wrel  = (const float*)d_in[3 + 3 * l + 0]; // (fout, fin)
        const float* brel  = (const float*)d_in[3 + 3 * l + 1]; // (fout,)
        const float* Wroot = (const float*)d_in[3 + 3 * l + 2]; // (fout, fin)

        float* outBuf = (l & 1) ? hB : hA;

        int rowTiles  = (int)(N / 16);              // exact
        int colTiles  = (fout + 15) / 16;
        int rowGroups = (rowTiles + 7) / 8;
        unsigned gblocks = (unsigned)rowGroups * (unsigned)colTiles;
        size_t shBytes = (size_t)(16 * pad4(fin) * 2 + 16) * sizeof(float);
        gemm_dual_wmma<<<gblocks, 256, shBytes, stream>>>(
            agg, Wrel, h, Wroot, brel,
            outBuf, (int)N, fout, fin, sIn, sOut, colTiles, /*relu=*/1);

        h = outBuf;
    }

    // ---- MLP: final (N,2) conv output (compact) viewed as (Nm, 396) ----
    int Nm = (int)((N * 2) / MLP_DIM);              // 4096 (multiple of 16)
    float* z1 = agg;                                 // reuse agg region
    float* z2 = agg + (long long)Nm * MLP_DIM;
    float* z3 = hA;                                  // last conv wrote hB

    const float* W1 = (const float*)d_in[33];
    const float* b1 = (const float*)d_in[34];
    const float* W2 = (const float*)d_in[35];
    const float* b2 = (const float*)d_in[36];
    const float* W3 = (const float*)d_in[37];
    const float* b3 = (const float*)d_in[38];

    int rowTiles  = (Nm + 15) / 16;                  // 256
    int colTiles  = (MLP_DIM + 15) / 16;             // 25
    int rowGroups = (rowTiles + 7) / 8;              // 32
    unsigned mblocks = (unsigned)rowGroups * (unsigned)colTiles;
    size_t mShBytes = (size_t)(16 * MLP_DIM + 16) * sizeof(float);  // 396 % 4 == 0

    gemm_dual_wmma<<<mblocks, 256, mShBytes, stream>>>(
        h,  W1, nullptr, nullptr, b1, z1, Nm, MLP_DIM, MLP_DIM, MLP_DIM, MLP_DIM, colTiles, 1);
    gemm_dual_wmma<<<mblocks, 256, mShBytes, stream>>>(
        z1, W2, nullptr, nullptr, b2, z2, Nm, MLP_DIM, MLP_DIM, MLP_DIM, MLP_DIM, colTiles, 1);
    gemm_dual_wmma<<<mblocks, 256, mShBytes, stream>>>(
        z2, W3, nullptr, nullptr, b3, z3, Nm, MLP_DIM, MLP_DIM, MLP_DIM, MLP_DIM, colTiles, 0);

    // ---- pairwise softmax -> d_out ----
    long long P = N;                                 // (Nm*396)/2 == N pairs
    unsigned pblocks = (unsigned)((P + 255) / 256);
    softmax2<<<pblocks, 256, 0, stream>>>(z3, (float*)d_out, P);
}